// CriticGNN_54202487276072
// MI455X (gfx1250) — compile-verified
//
#include <hip/hip_runtime.h>
#include <hip/hip_bf16.h>

typedef __attribute__((ext_vector_type(2))) float v2f;
typedef __attribute__((ext_vector_type(8))) float v8f;

// ---------------------------------------------------------------------------
// degree / norm preparation
// ---------------------------------------------------------------------------
__global__ void deg_init_k(float* __restrict__ deg, int n) {
    int i = blockIdx.x * blockDim.x + threadIdx.x;
    if (i < n) deg[i] = 1.0f;                       // self-loop contributes 1
}

__global__ void deg_accum_k(const int* __restrict__ dst, float* __restrict__ deg, int E) {
    int e = blockIdx.x * blockDim.x + threadIdx.x;
    if (e < E) atomicAdd(&deg[dst[e]], 1.0f);
}

__global__ void deg_finish_k(float* __restrict__ dinv, int n) {
    int i = blockIdx.x * blockDim.x + threadIdx.x;
    if (i < n) dinv[i] = rsqrtf(dinv[i]);           // deg >= 1 always (self-loop)
}

__global__ void edge_norm_k(const int* __restrict__ src, const int* __restrict__ dst,
                            const float* __restrict__ dinv, float* __restrict__ norm, int E) {
    int e = blockIdx.x * blockDim.x + threadIdx.x;
    if (e < E) norm[e] = dinv[src[e]] * dinv[dst[e]];
}

__global__ void zero_k(float* __restrict__ p, int n) {
    int i = blockIdx.x * blockDim.x + threadIdx.x;
    if (i < n) p[i] = 0.0f;
}

// ---------------------------------------------------------------------------
// fp32 node-feature GEMM on V_WMMA_F32_16X16X4_F32
//   in : [Ntiles*16, K] row-major, W : [K, NOUT] row-major, out : [Ntiles*16, NOUT]
// One wave computes one 16-row tile (all NOUT columns). N tiles of 16 cols,
// K tiles of 4. EXEC is full: the bounds check is wave-uniform.
// ---------------------------------------------------------------------------
template <int K, int NOUT>
__global__ void node_gemm_f32(const float* __restrict__ in, const float* __restrict__ W,
                              float* __restrict__ out, int nTiles) {
    int wave = (blockIdx.x * blockDim.x + threadIdx.x) >> 5;
    int lane = threadIdx.x & 31;
    if (wave >= nTiles) return;                      // uniform per wave

    const int row0 = wave * 16;
    const int m    = lane & 15;
    const int koff = (lane >> 4) << 1;               // 0 for lanes 0-15, 2 for 16-31
    const int n    = lane & 15;
    const float* arow = in + (size_t)(row0 + m) * K;

    #pragma unroll
    for (int nt = 0; nt < NOUT / 16; ++nt) {
        v8f c = {};
        #pragma unroll
        for (int s = 0; s < K / 4; ++s) {
            int kk = s * 4 + koff;
            v2f a, b;
            a.x = arow[kk];
            a.y = arow[kk + 1];
            b.x = W[(size_t)kk * NOUT + nt * 16 + n];
            b.y = W[(size_t)(kk + 1) * NOUT + nt * 16 + n];
            c = __builtin_amdgcn_wmma_f32_16x16x4_f32(false, a, false, b,
                                                      (short)0, c, false, false);
        }
        const int mbase = row0 + ((lane >> 4) << 3); // +8 for lanes 16-31
        #pragma unroll
        for (int v = 0; v < 8; ++v)
            out[(size_t)(mbase + v) * NOUT + nt * 16 + n] = c[v];
    }
}

// ---------------------------------------------------------------------------
// edge scatter: agg[dst, f] += h[src, f] * norm[e]   (one thread per (e,f))
// ---------------------------------------------------------------------------
template <int D>
__global__ void scatter_k(const int* __restrict__ src, const int* __restrict__ dst,
                          const float* __restrict__ norm, const float* __restrict__ h,
                          float* __restrict__ agg, int E) {
    int gid = blockIdx.x * blockDim.x + threadIdx.x;
    int e = gid / D;
    int f = gid - e * D;
    if (e >= E) return;
    float v = h[(size_t)src[e] * D + f] * norm[e];
    atomicAdd(&agg[(size_t)dst[e] * D + f], v);
}

// finish: add self-loop term + bias, optional ReLU, in place on agg
template <int D, bool RELU>
__global__ void finish_k(const float* __restrict__ h, const float* __restrict__ dinv,
                         const float* __restrict__ b, float* __restrict__ agg, int N) {
    int gid = blockIdx.x * blockDim.x + threadIdx.x;
    int i = gid / D;
    int f = gid - i * D;
    if (i >= N) return;
    float di = dinv[i];
    float v = agg[gid] + h[gid] * di * di + b[f];
    if (RELU) v = fmaxf(v, 0.0f);
    agg[gid] = v;
}

// ---------------------------------------------------------------------------
// pooling + head
// ---------------------------------------------------------------------------
__global__ void pool_feat_k(const float* __restrict__ h, const int* __restrict__ batch,
                            float* __restrict__ sums, int N) {
    int gid = blockIdx.x * blockDim.x + threadIdx.x;
    int i = gid >> 5;                                // D == 32
    int f = gid & 31;
    if (i < N) atomicAdd(&sums[batch[i] * 32 + f], h[gid]);
}

__global__ void pool_cnt_k(const int* __restrict__ batch, float* __restrict__ cnts, int N) {
    int i = blockIdx.x * blockDim.x + threadIdx.x;
    if (i < N) atomicAdd(&cnts[batch[i]], 1.0f);
}

__global__ void head_k(const float* __restrict__ sums, const float* __restrict__ cnts,
                       const float* __restrict__ action,
                       const float* __restrict__ Wp1, const float* __restrict__ bp1,
                       const float* __restrict__ Wp2, const float* __restrict__ bp2,
                       const float* __restrict__ Wp3, const float* __restrict__ bp3,
                       float* __restrict__ out) {
    int g = threadIdx.x;
    if (g >= 64) return;
    float inv = 1.0f / fmaxf(cnts[g], 1.0f);
    float mol[32];
    #pragma unroll
    for (int f = 0; f < 32; ++f) mol[f] = sums[g * 32 + f] * inv;

    float fp[64];
    for (int o = 0; o < 64; ++o) {
        float acc = bp1[o];
        #pragma unroll
        for (int i = 0; i < 32; ++i) acc += mol[i] * Wp1[i * 64 + o];
        fp[o] = fmaxf(acc, 0.0f);
    }
    float pol[10];
    for (int o = 0; o < 10; ++o) {
        float acc = bp2[o];
        for (int i = 0; i < 64; ++i) acc += fp[i] * Wp2[i * 10 + o];
        #pragma unroll
        for (int i = 0; i < 4; ++i) acc += action[g * 4 + i] * Wp2[(64 + i) * 10 + o];
        pol[o] = fmaxf(acc, 0.0f);                  // relu(pol)
    }
    float acc = bp3[0];
    #pragma unroll
    for (int o = 0; o < 10; ++o) acc += pol[o] * Wp3[o];
    out[g] = acc;
}

// ---------------------------------------------------------------------------
// orchestration
// ---------------------------------------------------------------------------
extern "C" void kernel_launch(void* const* d_in, const int* in_sizes, int n_in,
                              void* d_out, int out_size, void* d_ws, size_t ws_size,
                              hipStream_t stream) {
    const float* x      = (const float*)d_in[0];
    const int*   ei     = (const int*)d_in[1];
    const int*   batch  = (const int*)d_in[2];
    const float* action = (const float*)d_in[3];
    const float* W1 = (const float*)d_in[4];  const float* b1 = (const float*)d_in[5];
    const float* W2 = (const float*)d_in[6];  const float* b2 = (const float*)d_in[7];
    const float* W3 = (const float*)d_in[8];  const float* b3 = (const float*)d_in[9];
    const float* Wp1 = (const float*)d_in[10]; const float* bp1 = (const float*)d_in[11];
    const float* Wp2 = (const float*)d_in[12]; const float* bp2 = (const float*)d_in[13];
    const float* Wp3 = (const float*)d_in[14]; const float* bp3 = (const float*)d_in[15];
    float* out = (float*)d_out;

    const int N = in_sizes[0] / 128;   // 100000
    const int E = in_sizes[1] / 2;     // 1600000
    const int* srcI = ei;
    const int* dstI = ei + E;

    float* ws   = (float*)d_ws;
    float* bufA = ws;                          // N*64 : GEMM output (h)
    float* bufB = bufA + (size_t)N * 64;       // N*64 : aggregation / layer output
    float* dinv = bufB + (size_t)N * 64;       // N
    float* norm = dinv + N;                    // E
    float* sums = norm + E;                    // 64*32
    float* cnts = sums + 64 * 32;              // 64

    const int T = 256;
    auto cdiv = [](int a, int b) { return (a + b - 1) / b; };

    // degrees + per-edge norm (shared by all 3 layers)
    deg_init_k<<<cdiv(N, T), T, 0, stream>>>(dinv, N);
    deg_accum_k<<<cdiv(E, T), T, 0, stream>>>(dstI, dinv, E);
    deg_finish_k<<<cdiv(N, T), T, 0, stream>>>(dinv, N);
    edge_norm_k<<<cdiv(E, T), T, 0, stream>>>(srcI, dstI, dinv, norm, E);

    const int tiles = N / 16;                  // 6250 (N divisible by 16)
    const int gB = cdiv(tiles, T / 32);        // 8 waves per block

    // ---- layer 1: 128 -> 16, relu ----
    node_gemm_f32<128, 16><<<gB, T, 0, stream>>>(x, W1, bufA, tiles);
    zero_k<<<cdiv(N * 16, T), T, 0, stream>>>(bufB, N * 16);
    scatter_k<16><<<cdiv(E * 16, T), T, 0, stream>>>(srcI, dstI, norm, bufA, bufB, E);
    finish_k<16, true><<<cdiv(N * 16, T), T, 0, stream>>>(bufA, dinv, b1, bufB, N);

    // ---- layer 2: 16 -> 64, relu ----
    node_gemm_f32<16, 64><<<gB, T, 0, stream>>>(bufB, W2, bufA, tiles);
    zero_k<<<cdiv(N * 64, T), T, 0, stream>>>(bufB, N * 64);
    scatter_k<64><<<cdiv(E * 64, T), T, 0, stream>>>(srcI, dstI, norm, bufA, bufB, E);
    finish_k<64, true><<<cdiv(N * 64, T), T, 0, stream>>>(bufA, dinv, b2, bufB, N);

    // ---- layer 3: 64 -> 32, no relu ----
    node_gemm_f32<64, 32><<<gB, T, 0, stream>>>(bufB, W3, bufA, tiles);
    zero_k<<<cdiv(N * 32, T), T, 0, stream>>>(bufB, N * 32);
    scatter_k<32><<<cdiv(E * 32, T), T, 0, stream>>>(srcI, dstI, norm, bufA, bufB, E);
    finish_k<32, false><<<cdiv(N * 32, T), T, 0, stream>>>(bufA, dinv, b3, bufB, N);

    // ---- mean pool + MLP head ----
    zero_k<<<cdiv(64 * 32 + 64, T), T, 0, stream>>>(sums, 64 * 32 + 64);  // sums+cnts contiguous
    pool_feat_k<<<cdiv(N * 32, T), T, 0, stream>>>(bufB, batch, sums, N);
    pool_cnt_k<<<cdiv(N, T), T, 0, stream>>>(batch, cnts, N);
    head_k<<<1, 64, 0, stream>>>(sums, cnts, action, Wp1, bp1, Wp2, bp2, Wp3, bp3, out);
}